// GraphAttentionLayer_12472585028039
// MI455X (gfx1250) — compile-verified
//
#include <hip/hip_runtime.h>
#include <hip/hip_bf16.h>

typedef __attribute__((ext_vector_type(2))) float v2f;
typedef __attribute__((ext_vector_type(8))) float v8f;

namespace {
constexpr int B_   = 4;
constexpr int N_   = 4096;
constexpr int FIN  = 128;
constexpr int FOUT = 64;
constexpr int NH   = N_ / 2;          // 2048
constexpr float LALPHA = 0.2f;        // leaky-relu slope

__device__ __forceinline__ float lrelu(float x) { return x > 0.f ? x : LALPHA * x; }
} // namespace

// ---------------------------------------------------------------------------
// Kernel 1: h = inp @ W  via V_WMMA_F32_16X16X4_F32 (exact f32 path).
// Each block = 4 waves; wave w computes a 16x16 tile: rows [16*bid,16*bid+16),
// cols [16w, 16w+16). K loops 128/4 = 32 WMMA steps.
// A frag (16x4 f32): lanes 0-15 hold M=lane, K={0,1}; lanes 16-31 M=lane-16,
// K={2,3} (v0=K even base, v1=+1). B frag mirrors with K on lane halves.
// C/D (16x16 f32): VGPR v holds M=v (lanes 0-15) / M=8+v (lanes 16-31), N=lane%16.
// ---------------------------------------------------------------------------
__global__ __launch_bounds__(128) void gat_h_wmma(const float* __restrict__ inp,
                                                  const float* __restrict__ W,
                                                  float* __restrict__ h) {
  const int wave = threadIdx.x >> 5;
  const int lane = threadIdx.x & 31;
  const int r0   = blockIdx.x * 16;
  const int c0   = wave * 16;
  const int rsel = lane & 15;
  const int hi   = lane >> 4;      // 0 or 1
  const int kh   = hi << 1;        // K sub-offset 0 or 2

  const float* arow = inp + (size_t)(r0 + rsel) * FIN + kh;
  const float* bcol = W + (size_t)kh * FOUT + (c0 + rsel);

  v8f acc = {};
#pragma unroll
  for (int k = 0; k < FIN; k += 4) {
    v2f a, bf;
    a.x  = arow[k];
    a.y  = arow[k + 1];
    bf.x = bcol[(size_t)k * FOUT];
    bf.y = bcol[(size_t)(k + 1) * FOUT];
    acc = __builtin_amdgcn_wmma_f32_16x16x4_f32(false, a, false, bf,
                                                (short)0, acc, false, false);
  }

  float* hout = h + (size_t)(r0 + hi * 8) * FOUT + c0 + rsel;
#pragma unroll
  for (int v = 0; v < 8; ++v) hout[(size_t)v * FOUT] = acc[v];
}

// ---------------------------------------------------------------------------
// Kernel 2: per-batch attention statistics, entirely in LDS (48.5 KB).
//   s1,s2 -> t[n]=s1+s2, u[j]=s1[2j%N]+s2[(2j+1)%N]
//   M = max(L(t),L(u));  S_even/S_odd = sum exp(L(t[even/odd]) - M)
//   D[j] = (j<NH ? S_even : S_odd) + NH*exp(L(u[j])-M)
//   P = sum_{j<NH}  h[j,:]/D[j];  Q = sum_{j>=NH} h[j,:]/D[j]
//   R = sum_j (exp(L(u[j])-M)/D[j]) * h[j,:]
//   ee[i]=exp(L(t[2i])-M), eo[i]=exp(L(t[2i+1])-M)  (written to global)
// ---------------------------------------------------------------------------
__global__ __launch_bounds__(1024) void gat_stats(const float* __restrict__ h,
                                                  const float* __restrict__ a,
                                                  float* __restrict__ ee,
                                                  float* __restrict__ eo,
                                                  float* __restrict__ P,
                                                  float* __restrict__ Q,
                                                  float* __restrict__ R) {
  __shared__ float A0[N_];          // s1 -> t -> e_t -> reduce scratch
  __shared__ float A1[N_];          // s2 -> reduce scratch -> 1/D
  __shared__ float A2[N_];          // u  -> exp(L(u)-M)/D
  __shared__ float ash[2 * FOUT];
  __shared__ float bc[4];

  const int b   = blockIdx.x;
  const int tid = threadIdx.x;
  const float* hb = h + (size_t)b * N_ * FOUT;

  if (tid < 2 * FOUT) ash[tid] = a[tid];
  __syncthreads();

  // s1, s2 (each thread: 4 rows of 64-length dots)
  for (int k = 0; k < 4; ++k) {
    const int n = tid + k * 1024;
    const float* hr = hb + (size_t)n * FOUT;
    float d1 = 0.f, d2 = 0.f;
#pragma unroll
    for (int f = 0; f < FOUT; ++f) {
      const float hv = hr[f];
      d1 += hv * ash[f];
      d2 += hv * ash[FOUT + f];
    }
    A0[n] = d1;
    A1[n] = d2;
  }
  __syncthreads();

  // u[j]
  for (int k = 0; k < 4; ++k) {
    const int j = tid + k * 1024;
    A2[j] = A0[(2 * j) & (N_ - 1)] + A1[(2 * j + 1) & (N_ - 1)];
  }
  __syncthreads();

  // t[n] in place
  for (int k = 0; k < 4; ++k) {
    const int n = tid + k * 1024;
    A0[n] = A0[n] + A1[n];
  }
  __syncthreads();

  // global max M over L(t), L(u)
  float pm = -3.4e38f;
  for (int k = 0; k < 4; ++k) {
    const int n = tid + k * 1024;
    pm = fmaxf(pm, fmaxf(lrelu(A0[n]), lrelu(A2[n])));
  }
  A1[tid] = pm;
  __syncthreads();
  for (int s = 512; s > 0; s >>= 1) {
    if (tid < s) A1[tid] = fmaxf(A1[tid], A1[tid + s]);
    __syncthreads();
  }
  if (tid == 0) bc[0] = A1[0];
  __syncthreads();
  const float M = bc[0];

  // e_t = exp(L(t)-M); write ee/eo; partial sums by parity (n parity == tid parity)
  float pe = 0.f;
  for (int k = 0; k < 4; ++k) {
    const int n = tid + k * 1024;
    const float e = expf(lrelu(A0[n]) - M);
    A0[n] = e;
    if (n & 1) eo[b * NH + (n >> 1)] = e;
    else       ee[b * NH + (n >> 1)] = e;
    pe += e;
  }
  A1[tid] = (tid & 1) ? 0.f : pe;
  __syncthreads();
  for (int s = 512; s > 0; s >>= 1) { if (tid < s) A1[tid] += A1[tid + s]; __syncthreads(); }
  if (tid == 0) bc[1] = A1[0];
  __syncthreads();
  A1[tid] = (tid & 1) ? pe : 0.f;
  __syncthreads();
  for (int s = 512; s > 0; s >>= 1) { if (tid < s) A1[tid] += A1[tid + s]; __syncthreads(); }
  if (tid == 0) bc[2] = A1[0];
  __syncthreads();
  const float Se = bc[1];
  const float So = bc[2];
  __syncthreads();

  // 1/D[j] and exp(L(u)-M)/D[j]
  for (int k = 0; k < 4; ++k) {
    const int j = tid + k * 1024;
    const float ew  = expf(lrelu(A2[j]) - M);
    const float D   = ((j < NH) ? Se : So) + (float)NH * ew;
    const float inv = 1.f / D;
    A1[j] = inv;
    A2[j] = ew * inv;
  }
  __syncthreads();

  // P, Q, R: 16 groups x 64 features
  const int f = tid & (FOUT - 1);
  const int g = tid >> 6;
  float pP = 0.f, pQ = 0.f, pR = 0.f;
  for (int j = g; j < N_; j += 16) {
    const float hv = hb[(size_t)j * FOUT + f];
    const float w1 = A1[j];
    if (j < NH) pP += hv * w1; else pQ += hv * w1;
    pR += hv * A2[j];
  }
  A0[tid] = pP;
  __syncthreads();
  for (int s = 8; s > 0; s >>= 1) { if (g < s) A0[tid] += A0[tid + s * 64]; __syncthreads(); }
  if (g == 0) P[b * FOUT + f] = A0[tid];
  __syncthreads();
  A0[tid] = pQ;
  __syncthreads();
  for (int s = 8; s > 0; s >>= 1) { if (g < s) A0[tid] += A0[tid + s * 64]; __syncthreads(); }
  if (g == 0) Q[b * FOUT + f] = A0[tid];
  __syncthreads();
  A0[tid] = pR;
  __syncthreads();
  for (int s = 8; s > 0; s >>= 1) { if (g < s) A0[tid] += A0[tid + s * 64]; __syncthreads(); }
  if (g == 0) R[b * FOUT + f] = A0[tid];
}

// ---------------------------------------------------------------------------
// Kernel 3: out[b,n,f] = elu( n<NH ? ee[n]*P[f] + eo[n]*Q[f] : R[f] )
// ---------------------------------------------------------------------------
__global__ __launch_bounds__(256) void gat_out(const float* __restrict__ ee,
                                               const float* __restrict__ eo,
                                               const float* __restrict__ P,
                                               const float* __restrict__ Q,
                                               const float* __restrict__ R,
                                               float* __restrict__ out) {
  const int idx = blockIdx.x * 256 + threadIdx.x;
  const int f = idx & (FOUT - 1);
  const int n = (idx >> 6) & (N_ - 1);
  const int b = idx >> 18;                    // N*FOUT = 2^18
  float hp;
  if (n < NH) hp = ee[b * NH + n] * P[b * FOUT + f] + eo[b * NH + n] * Q[b * FOUT + f];
  else        hp = R[b * FOUT + f];
  out[idx] = hp > 0.f ? hp : (expf(hp) - 1.f);
}

extern "C" void kernel_launch(void* const* d_in, const int* in_sizes, int n_in,
                              void* d_out, int out_size, void* d_ws, size_t ws_size,
                              hipStream_t stream) {
  (void)in_sizes; (void)n_in; (void)out_size; (void)ws_size;
  const float* inp = (const float*)d_in[0];   // (B, N, FIN)
  const float* W   = (const float*)d_in[1];   // (FIN, FOUT)
  const float* a   = (const float*)d_in[2];   // (2*FOUT, 1)
  float* out = (float*)d_out;                 // (B, N, FOUT)

  float* h  = (float*)d_ws;                   // B*N*FOUT floats (4 MB)
  float* ee = h  + (size_t)B_ * N_ * FOUT;    // B*NH
  float* eo = ee + (size_t)B_ * NH;           // B*NH
  float* P  = eo + (size_t)B_ * NH;           // B*FOUT
  float* Q  = P  + (size_t)B_ * FOUT;         // B*FOUT
  float* R  = Q  + (size_t)B_ * FOUT;         // B*FOUT

  gat_h_wmma<<<dim3((B_ * N_) / 16), dim3(128), 0, stream>>>(inp, W, h);
  gat_stats<<<dim3(B_), dim3(1024), 0, stream>>>(h, a, ee, eo, P, Q, R);
  gat_out<<<dim3((B_ * N_ * FOUT) / 256), dim3(256), 0, stream>>>(ee, eo, P, Q, R, out);
}